// UNetModel50K_quan_66975720014021
// MI455X (gfx1250) — compile-verified
//
#include <hip/hip_runtime.h>
#include <math.h>

typedef float v2f __attribute__((ext_vector_type(2)));
typedef float v8f __attribute__((ext_vector_type(8)));

#define QMAXF 32767.0f
#define SLOPE 0.1015625f

// ---------------------------------------------------------------------------
// scale-slot init
// ---------------------------------------------------------------------------
__global__ void zero_scales_kernel(unsigned* s, int n) {
    int i = blockIdx.x * blockDim.x + threadIdx.x;
    if (i < n) s[i] = 0u;
}

// ---------------------------------------------------------------------------
// per-tensor fake-quant of weights; pw is padded to [Mpad x Kpad]
// row-major (M = cout rows, K = cin cols, zero padding) for WMMA A loads.
// grid.x == 2: block 0 -> depthwise weights, block 1 -> pointwise weights
// ---------------------------------------------------------------------------
__global__ void quant_weights_kernel(const float* __restrict__ dw,
                                     const float* __restrict__ pw,
                                     float* __restrict__ dwq,
                                     float* __restrict__ pwq,
                                     int cin, int cout, int Kpad, int Mpad) {
    __shared__ float red[256];
    int tid = threadIdx.x;
    if (blockIdx.x == 0) {
        int n = cin * 9;
        float m = 0.f;
        for (int i = tid; i < n; i += blockDim.x) m = fmaxf(m, fabsf(dw[i]));
        red[tid] = m; __syncthreads();
        for (int s = 128; s > 0; s >>= 1) {
            if (tid < s) red[tid] = fmaxf(red[tid], red[tid + s]);
            __syncthreads();
        }
        float scale = fmaxf(red[0] / QMAXF, 1e-12f);
        for (int i = tid; i < n; i += blockDim.x) {
            float q = rintf(dw[i] / scale);
            q = fminf(fmaxf(q, -QMAXF), QMAXF);
            dwq[i] = q * scale;
        }
    } else {
        int n = cout * cin;
        float m = 0.f;
        for (int i = tid; i < n; i += blockDim.x) m = fmaxf(m, fabsf(pw[i]));
        red[tid] = m; __syncthreads();
        for (int s = 128; s > 0; s >>= 1) {
            if (tid < s) red[tid] = fmaxf(red[tid], red[tid + s]);
            __syncthreads();
        }
        float scale = fmaxf(red[0] / QMAXF, 1e-12f);
        int tot = Mpad * Kpad;
        for (int i = tid; i < tot; i += blockDim.x) {
            int mr = i / Kpad, k = i % Kpad;
            float v = 0.f;
            if (mr < cout && k < cin) {
                float q = rintf(pw[mr * cin + k] / scale);
                q = fminf(fmaxf(q, -QMAXF), QMAXF);
                v = q * scale;
            }
            pwq[i] = v;
        }
    }
}

// ---------------------------------------------------------------------------
// depthwise 3x3, pad=1, stride=1.
// input layout  [N][Cin ][H][W]  (unpadded)
// output layout [N][Kpad][H][W]  (K-padded for the WMMA consumer; pad
//                                 channels written as zero -> no guards later)
// ---------------------------------------------------------------------------
__global__ void dw3x3_kernel(const float* __restrict__ in,
                             const float* __restrict__ w,
                             float* __restrict__ out,
                             int Cin, int Kpad, int H, int W, long long total) {
    long long idx = (long long)blockIdx.x * blockDim.x + threadIdx.x;
    long long stride = (long long)gridDim.x * blockDim.x;
    for (; idx < total; idx += stride) {
        int x = (int)(idx % W);
        long long t = idx / W;
        int y = (int)(t % H); t /= H;
        int c = (int)(t % Kpad);
        long long n = t / Kpad;
        if (c >= Cin) { out[idx] = 0.f; continue; }
        const float* wp = w + c * 9;
        const float* ip = in + (n * Cin + c) * (long long)H * W;
        float acc = 0.f;
#pragma unroll
        for (int ky = 0; ky < 3; ky++) {
            int yy = y + ky - 1;
            if ((unsigned)yy >= (unsigned)H) continue;
#pragma unroll
            for (int kx = 0; kx < 3; kx++) {
                int xx = x + kx - 1;
                if ((unsigned)xx >= (unsigned)W) continue;
                acc += ip[(long long)yy * W + xx] * wp[ky * 3 + kx];
            }
        }
        out[idx] = acc;
    }
}

// ---------------------------------------------------------------------------
// pointwise 1x1 conv as fp32 WMMA GEMM (V_WMMA_F32_16X16X4_F32):
//   D[16 cout x 16 pix] += A[16x4] * B[4x16]
// A (weights): lanes 0-15 = rows M with K,K+1 ; lanes 16-31 = K+2,K+3
// B (acts):    VGPR0 lanes 0-15 = row K, lanes 16-31 = row K+1;
//              VGPR1 lanes 0-15 = row K+2, lanes 16-31 = row K+3
// Each wave computes 4 N-tiles (64 pixels) to reuse each A load 4x.
// Branch-free K loop (activations are K-padded with zeros).
// Fused: + bias, LeakyReLU, block max-abs -> atomic max (for fake-quant).
// grid: (P/512, Mpad/16, batch=8); 256 threads = 8 waves.
// ---------------------------------------------------------------------------
__global__ void pw_wmma_kernel(const float* __restrict__ act,
                               const float* __restrict__ wq,
                               const float* __restrict__ bias,
                               float* __restrict__ out,
                               int Kpad, int Cout, int P, int relu,
                               unsigned* __restrict__ maxslot) {
    __shared__ unsigned smax;
    if (threadIdx.x == 0) smax = 0u;
    __syncthreads();

    int lane = threadIdx.x & 31;
    int wave = threadIdx.x >> 5;
    int hi = lane >> 4;   // lane half: 0 or 1
    int lo = lane & 15;
    int nbase = (blockIdx.x * 8 + wave) * 64;    // 4 tiles of 16 pixels
    int m_base = blockIdx.y * 16;
    long long img = blockIdx.z;

    const float* A = act + img * (long long)Kpad * P;       // padded activations
    const float* aptr = wq + (m_base + lo) * Kpad + 2 * hi; // weight row, K pair
    const float* b0 = A + (long long)hi * P + nbase + lo;   // rows K+hi
    const float* b1 = b0 + 2 * (long long)P;                // rows K+2+hi
    long long kstep = 4 * (long long)P;

    v8f acc0 = {}, acc1 = {}, acc2 = {}, acc3 = {};
    for (int k = 0; k < Kpad; k += 4) {
        v2f a = *(const v2f*)aptr;   // 8B-aligned: Kpad%4==0
        __builtin_prefetch(b0 + kstep, 0, 0);  // global_prefetch_b8: next K rows
        v2f bb;
        bb.x = b0[0];  bb.y = b1[0];
        acc0 = __builtin_amdgcn_wmma_f32_16x16x4_f32(false, a, false, bb, (short)0, acc0, false, false);
        bb.x = b0[16]; bb.y = b1[16];
        acc1 = __builtin_amdgcn_wmma_f32_16x16x4_f32(false, a, false, bb, (short)0, acc1, false, false);
        bb.x = b0[32]; bb.y = b1[32];
        acc2 = __builtin_amdgcn_wmma_f32_16x16x4_f32(false, a, false, bb, (short)0, acc2, false, false);
        bb.x = b0[48]; bb.y = b1[48];
        acc3 = __builtin_amdgcn_wmma_f32_16x16x4_f32(false, a, false, bb, (short)0, acc3, false, false);
        aptr += 4; b0 += kstep; b1 += kstep;
    }

    float* O = out + img * (long long)Cout * P;
    float lmax = 0.f;
#pragma unroll
    for (int r = 0; r < 8; r++) {
        int mrow = m_base + r + hi * 8;   // C/D layout: VGPR r -> M=r / M=r+8
        if (mrow < Cout) {
            float bsv = bias[mrow];
            float v0 = acc0[r] + bsv, v1 = acc1[r] + bsv;
            float v2 = acc2[r] + bsv, v3 = acc3[r] + bsv;
            if (relu) {
                v0 = (v0 >= 0.f) ? v0 : SLOPE * v0;
                v1 = (v1 >= 0.f) ? v1 : SLOPE * v1;
                v2 = (v2 >= 0.f) ? v2 : SLOPE * v2;
                v3 = (v3 >= 0.f) ? v3 : SLOPE * v3;
            }
            lmax = fmaxf(lmax, fmaxf(fmaxf(fabsf(v0), fabsf(v1)),
                                     fmaxf(fabsf(v2), fabsf(v3))));
            long long rowoff = (long long)mrow * P + nbase + lo;
            O[rowoff]      = v0;
            O[rowoff + 16] = v1;
            O[rowoff + 32] = v2;
            O[rowoff + 48] = v3;
        }
    }
    if (relu) {
        atomicMax(&smax, __float_as_uint(lmax));  // |v| >= 0: uint order == float order
        __syncthreads();
        if (threadIdx.x == 0) atomicMax(maxslot, smax);
    }
}

// ---------------------------------------------------------------------------
// in-place activation fake-quant using the max-abs slot produced above
// ---------------------------------------------------------------------------
__global__ void quantize_act_kernel(float* __restrict__ buf, long long total,
                                    const unsigned* __restrict__ maxslot) {
    float s = fmaxf(__uint_as_float(*maxslot) / QMAXF, 1e-12f);
    long long idx = (long long)blockIdx.x * blockDim.x + threadIdx.x;
    long long stride = (long long)gridDim.x * blockDim.x;
    for (; idx < total; idx += stride) {
        float v = buf[idx];
        float q = rintf(v / s);
        q = fminf(fmaxf(q, -QMAXF), QMAXF);
        buf[idx] = q * s;
    }
}

// ---------------------------------------------------------------------------
// 2x2 max pool, stride 2
// ---------------------------------------------------------------------------
__global__ void maxpool2_kernel(const float* __restrict__ in, float* __restrict__ out,
                                int H, int W, long long totalOut) {
    long long idx = (long long)blockIdx.x * blockDim.x + threadIdx.x;
    long long stride = (long long)gridDim.x * blockDim.x;
    int Wo = W >> 1, Ho = H >> 1;
    for (; idx < totalOut; idx += stride) {
        int x = (int)(idx % Wo);
        long long t = idx / Wo;
        int y = (int)(t % Ho);
        long long plane = t / Ho;  // n*C + c
        const float* ip = in + plane * (long long)H * W;
        int y2 = y * 2, x2 = x * 2;
        float m = fmaxf(fmaxf(ip[(long long)y2 * W + x2], ip[(long long)y2 * W + x2 + 1]),
                        fmaxf(ip[(long long)(y2 + 1) * W + x2], ip[(long long)(y2 + 1) * W + x2 + 1]));
        out[idx] = m;
    }
}

// ---------------------------------------------------------------------------
// out = concat([nearest-2x-up(x), skip], channel dim), NCHW
// ---------------------------------------------------------------------------
__global__ void up_concat_kernel(const float* __restrict__ xs,
                                 const float* __restrict__ skip,
                                 float* __restrict__ out,
                                 int C1, int C2, int Ho, int Wo, long long totalOut) {
    long long idx = (long long)blockIdx.x * blockDim.x + threadIdx.x;
    long long stride = (long long)gridDim.x * blockDim.x;
    int C = C1 + C2;
    for (; idx < totalOut; idx += stride) {
        int xx = (int)(idx % Wo);
        long long t = idx / Wo;
        int yy = (int)(t % Ho); t /= Ho;
        int c = (int)(t % C);
        long long n = t / C;
        float v;
        if (c < C1) {
            int Hi = Ho >> 1, Wi = Wo >> 1;
            v = xs[((n * C1 + c) * (long long)Hi + (yy >> 1)) * Wi + (xx >> 1)];
        } else {
            v = skip[((n * C2 + (c - C1)) * (long long)Ho + yy) * Wo + xx];
        }
        out[idx] = v;
    }
}

// ---------------------------------------------------------------------------
// host orchestration
// ---------------------------------------------------------------------------
struct Spec { int cin, cout; };
static const Spec SPECS[20] = {
    {3, 16},  {16, 16},   // enc1
    {16, 32}, {32, 32},   // enc2
    {32, 32}, {32, 48},   // enc3
    {48, 48}, {48, 48},   // enc4
    {48, 64}, {64, 64},   // mid
    {64, 64}, {64, 64},   // decm
    {112, 48}, {48, 48},  // dec1
    {96, 48}, {48, 32},   // dec2
    {64, 32}, {32, 32},   // dec3
    {48, 16}, {16, 2},    // dec4
};

extern "C" void kernel_launch(void* const* d_in, const int* in_sizes, int n_in,
                              void* d_out, int out_size, void* d_ws, size_t ws_size,
                              hipStream_t stream) {
    (void)in_sizes; (void)n_in; (void)out_size; (void)ws_size;
    const float* x = (const float*)d_in[0];
    char* ws = (char*)d_ws;

    // --- workspace layout (bump allocator, deterministic) ---
    size_t cur = 0;
    auto alloc = [&](size_t bytes) -> size_t {
        size_t o = cur;
        cur += (bytes + 255) & ~(size_t)255;
        return o;
    };
    size_t off_scales = alloc(64 * sizeof(unsigned));
    size_t off_dwq[20], off_pwq[20];
    int KpadA[20], MpadA[20];
    for (int i = 0; i < 20; i++) {
        int cin = SPECS[i].cin, cout = SPECS[i].cout;
        KpadA[i] = (cin + 3) & ~3;
        MpadA[i] = (cout + 15) & ~15;
        off_dwq[i] = alloc((size_t)cin * 9 * sizeof(float));
        off_pwq[i] = alloc((size_t)KpadA[i] * MpadA[i] * sizeof(float));
    }
    const size_t BUF_FLOATS = (size_t)8 * 48 * 512 * 512;   // biggest live tensor
    size_t off_bufA = alloc(BUF_FLOATS * sizeof(float));
    size_t off_bufB = alloc(BUF_FLOATS * sizeof(float));
    size_t off_s1 = alloc((size_t)8 * 16 * 512 * 512 * sizeof(float));
    size_t off_s2 = alloc((size_t)8 * 32 * 256 * 256 * sizeof(float));
    size_t off_s3 = alloc((size_t)8 * 48 * 128 * 128 * sizeof(float));
    size_t off_s4 = alloc((size_t)8 * 48 * 64 * 64 * sizeof(float));

    float* bufA = (float*)(ws + off_bufA);
    float* bufB = (float*)(ws + off_bufB);
    float* S1 = (float*)(ws + off_s1);
    float* S2 = (float*)(ws + off_s2);
    float* S3 = (float*)(ws + off_s3);
    float* S4 = (float*)(ws + off_s4);
    unsigned* scales = (unsigned*)(ws + off_scales);

    // --- weight fake-quant (tiny; recomputed every call, deterministic) ---
    zero_scales_kernel<<<1, 64, 0, stream>>>(scales, 64);
    for (int i = 0; i < 20; i++) {
        const float* dw = (const float*)d_in[1 + 3 * i + 0];
        const float* pw = (const float*)d_in[1 + 3 * i + 1];
        quant_weights_kernel<<<2, 256, 0, stream>>>(
            dw, pw, (float*)(ws + off_dwq[i]), (float*)(ws + off_pwq[i]),
            SPECS[i].cin, SPECS[i].cout, KpadA[i], MpadA[i]);
    }

    auto run_dw = [&](int i, const float* in, float* out, int H, int W) {
        long long total = (long long)8 * KpadA[i] * H * W;   // K-padded output
        int blocks = (int)((total + 255) / 256);
        dw3x3_kernel<<<blocks, 256, 0, stream>>>(
            in, (const float*)(ws + off_dwq[i]), out, SPECS[i].cin, KpadA[i], H, W, total);
    };
    auto run_pw = [&](int i, const float* in, float* out, int H, int W, int relu) {
        int P = H * W;                      // divisible by 512 for all levels here
        dim3 grid(P / 512, MpadA[i] / 16, 8);
        const float* bias = (const float*)d_in[1 + 3 * i + 2];
        pw_wmma_kernel<<<grid, 256, 0, stream>>>(
            in, (const float*)(ws + off_pwq[i]), bias, out,
            KpadA[i], SPECS[i].cout, P, relu, scales + i);
    };
    auto run_q = [&](int i, float* buf, int H, int W) {
        long long total = (long long)8 * SPECS[i].cout * H * W;
        int blocks = (int)((total + 255) / 256);
        quantize_act_kernel<<<blocks, 256, 0, stream>>>(buf, total, scales + i);
    };
    auto ds = [&](int i, const float* in, float* tmp, float* out, int H, int W, int relu) {
        run_dw(i, in, tmp, H, W);
        run_pw(i, tmp, out, H, W, relu);
        if (relu) run_q(i, out, H, W);
    };
    auto pool = [&](const float* in, float* out, int C, int H, int W) {
        long long total = (long long)8 * C * (H / 2) * (W / 2);
        maxpool2_kernel<<<(int)((total + 255) / 256), 256, 0, stream>>>(in, out, H, W, total);
    };
    auto upcat = [&](const float* xs, const float* skip, float* out,
                     int C1, int C2, int Ho, int Wo) {
        long long total = (long long)8 * (C1 + C2) * Ho * Wo;
        up_concat_kernel<<<(int)((total + 255) / 256), 256, 0, stream>>>(
            xs, skip, out, C1, C2, Ho, Wo, total);
    };

    // --- UNet ---
    ds(0, x, bufA, bufB, 512, 512, 1);          // enc1_b1 -> B (16@512)
    ds(1, bufB, bufA, S1, 512, 512, 1);         // enc1_b2 -> S1 (skip)
    pool(S1, bufA, 16, 512, 512);               // A: 16@256
    ds(2, bufA, bufB, bufA, 256, 256, 1);       // enc2_b1 -> A (32@256)
    ds(3, bufA, bufB, S2, 256, 256, 1);         // enc2_b2 -> S2
    pool(S2, bufA, 32, 256, 256);               // A: 32@128
    ds(4, bufA, bufB, bufA, 128, 128, 1);       // enc3_b1 -> A (32@128)
    ds(5, bufA, bufB, S3, 128, 128, 1);         // enc3_b2 -> S3 (48@128)
    pool(S3, bufA, 48, 128, 128);               // A: 48@64
    ds(6, bufA, bufB, bufA, 64, 64, 1);         // enc4_b1
    ds(7, bufA, bufB, S4, 64, 64, 1);           // enc4_b2 -> S4 (48@64)
    pool(S4, bufA, 48, 64, 64);                 // A: 48@32
    ds(8, bufA, bufB, bufA, 32, 32, 1);         // mid_b1 (64@32)
    ds(9, bufA, bufB, bufA, 32, 32, 1);         // mid_b2
    ds(10, bufA, bufB, bufA, 32, 32, 1);        // decm_b1
    ds(11, bufA, bufB, bufA, 32, 32, 1);        // decm_b2 (64@32)
    upcat(bufA, S4, bufB, 64, 48, 64, 64);      // B: 112@64
    ds(12, bufB, bufA, bufB, 64, 64, 1);        // dec1_b1 -> B (48@64)
    ds(13, bufB, bufA, bufB, 64, 64, 1);        // dec1_b2
    upcat(bufB, S3, bufA, 48, 48, 128, 128);    // A: 96@128
    ds(14, bufA, bufB, bufA, 128, 128, 1);      // dec2_b1 -> A (48@128)
    ds(15, bufA, bufB, bufA, 128, 128, 1);      // dec2_b2 -> A (32@128)
    upcat(bufA, S2, bufB, 32, 32, 256, 256);    // B: 64@256
    ds(16, bufB, bufA, bufB, 256, 256, 1);      // dec3_b1 -> B (32@256)
    ds(17, bufB, bufA, bufB, 256, 256, 1);      // dec3_b2
    upcat(bufB, S1, bufA, 32, 16, 512, 512);    // A: 48@512
    ds(18, bufA, bufB, bufA, 512, 512, 1);      // dec4_b1 -> A (16@512)
    ds(19, bufA, bufB, (float*)d_out, 512, 512, 0);  // dec4_b2 -> out (2@512)
}